// TimingPropagation_35622458753425
// MI455X (gfx1250) — compile-verified
//
#include <hip/hip_runtime.h>

// Timing-LUT interpolation for MI455X (gfx1250).
// Roofline: ~60 FLOP vs ~104 B per eval -> pure gather workload.
// The 164 MB LUT set (values/tables/dims) is L2-resident (192 MB global L2);
// streaming arrays are marked non-temporal so they don't evict it.

#define LUT_EPS 1e-12f

__global__ __launch_bounds__(256) void TimingPropagation_35622458753425_kernel(
    const float* __restrict__ input_trans,
    const float* __restrict__ output_caps,
    const int*   __restrict__ arc_idxs,
    const float* __restrict__ values,      // [N, 64]
    const float* __restrict__ trans_table, // [N, 8]
    const float* __restrict__ cap_table,   // [N, 8]
    const int*   __restrict__ dims,        // [N, 2]
    float*       __restrict__ out,
    int n)
{
    int i = blockIdx.x * blockDim.x + threadIdx.x;
    if (i >= n) return;

    // Streaming (zero-reuse) accesses: non-temporal so the L2 keeps the LUTs.
    const int a = __builtin_nontemporal_load(arc_idxs + i);
    const float* __restrict__ vrow = values + (size_t)a * 64;

    // Dependent gathers (v00..v11) can't issue until searchsorted finishes;
    // pull the 256B values row toward us now (gfx1250 global_prefetch_b8).
    __builtin_prefetch(vrow, 0, 3);
    __builtin_prefetch(vrow + 32, 0, 3);

    const float xt = __builtin_nontemporal_load(input_trans + i);
    const float xc = __builtin_nontemporal_load(output_caps + i);

    // 32B-aligned table rows -> two GLOBAL_LOAD_B128 each (L2-resident reuse,
    // regular-temporal).
    const float4* tp = (const float4*)(trans_table + (size_t)a * 8);
    const float4* cp = (const float4*)(cap_table   + (size_t)a * 8);
    float4 tA = tp[0], tB = tp[1];
    float4 cA = cp[0], cB = cp[1];
    int2 d = ((const int2*)dims)[a];
    const int td = d.x, cd = d.y;

    float tt[8] = {tA.x, tA.y, tA.z, tA.w, tB.x, tB.y, tB.z, tB.w};
    float cc[8] = {cA.x, cA.y, cA.z, cA.w, cB.x, cB.y, cB.z, cB.w};

    // searchsorted(side='right') over the full padded 8-entry row:
    // count of entries <= x. Branchless: 8 compares + adds.
    int ss_t = 0, ss_c = 0;
#pragma unroll
    for (int j = 0; j < 8; ++j) {
        ss_t += (tt[j] <= xt) ? 1 : 0;
        ss_c += (cc[j] <= xc) ? 1 : 0;
    }

    // Sequential clamps exactly as reference: clamp(min=1) then clamp(max=dim-1).
    const int max_t = max(td - 1, 0);
    const int max_c = max(cd - 1, 0);
    const int t_hi = min(max(ss_t, 1), max_t);
    const int c_hi = min(max(ss_c, 1), max_c);
    const int t_lo = max(t_hi - 1, 0);
    const int c_lo = max(c_hi - 1, 0);

    // Register-resident dynamic select (lowers to v_cndmask chains, no scratch).
    float t0 = tt[0], t1 = tt[0], c0 = cc[0], c1 = cc[0];
#pragma unroll
    for (int j = 1; j < 8; ++j) {
        t0 = (t_lo == j) ? tt[j] : t0;
        t1 = (t_hi == j) ? tt[j] : t1;
        c0 = (c_lo == j) ? cc[j] : c0;
        c1 = (c_hi == j) ? cc[j] : c1;
    }

    const float ti = t1 - t0;
    const float ci = c1 - c0;
    const bool t_deg = fabsf(ti) < LUT_EPS;
    const bool c_deg = fabsf(ci) < LUT_EPS;

    const float xtc = fminf(fmaxf(xt, t0), t1);
    const float xcc = fminf(fmaxf(xc, c0), c1);
    const float ti_s = t_deg ? LUT_EPS : ti;
    const float ci_s = c_deg ? LUT_EPS : ci;
    const float ft = fminf(fmaxf((xtc - t0) / ti_s, 0.0f), 1.0f);
    const float fc = fminf(fmaxf((xcc - c0) / ci_s, 0.0f), 1.0f);

    // Unified index formula: with cd==1 it degenerates to vals[t_lo]/vals[t_hi]
    // (1D-trans path); with td==1 to vals[c_lo]/vals[c_hi] (1D-cap); with both
    // ==1 to vals[0] (scalar). Max index = 7*8+7 = 63, always in-bounds.
    const int i00 = t_lo * cd + c_lo;
    const int i01 = t_lo * cd + c_hi;
    const int i10 = t_hi * cd + c_lo;
    const int i11 = t_hi * cd + c_hi;
    const float v00 = vrow[i00];
    const float v01 = vrow[i01];
    const float v10 = vrow[i10];
    const float v11 = vrow[i11];

    const float wa = (t1 - xtc) * (c1 - xcc);
    const float wb = (t1 - xtc) * (xcc - c0);
    const float wc = (xtc - t0) * (c1 - xcc);
    const float wd = (xtc - t0) * (xcc - c0);
    const float bil = (v00 * wa + v01 * wb + v10 * wc + v11 * wd) / (ti_s * ci_s);

    // Same nesting as reference's jnp.where chain; covers all four modes.
    const float v2d = (t_deg & c_deg) ? v00
                    : (t_deg ? v00 + (v01 - v00) * fc
                    : (c_deg ? v00 + (v10 - v00) * ft
                    : bil));

    const bool valid = (td > 0) & (cd > 0);
    __builtin_nontemporal_store(valid ? v2d : 0.0f, out + i);
}

extern "C" void kernel_launch(void* const* d_in, const int* in_sizes, int n_in,
                              void* d_out, int out_size, void* d_ws, size_t ws_size,
                              hipStream_t stream) {
    const float* input_trans = (const float*)d_in[0];
    const float* output_caps = (const float*)d_in[1];
    const int*   arc_idxs    = (const int*)  d_in[2];
    const float* values      = (const float*)d_in[3];
    const float* trans_tbl   = (const float*)d_in[4];
    const float* cap_tbl     = (const float*)d_in[5];
    const int*   dims        = (const int*)  d_in[6];
    float* out = (float*)d_out;

    const int n = in_sizes[0];                 // B = 2,000,000 evaluations
    const int block = 256;                     // 8 wave32 per block
    const int grid = (n + block - 1) / block;

    TimingPropagation_35622458753425_kernel<<<grid, block, 0, stream>>>(
        input_trans, output_caps, arc_idxs, values, trans_tbl, cap_tbl, dims,
        out, n);
}